// SelfOrganizingMap_8985071583883
// MI455X (gfx1250) — compile-verified
//
#include <hip/hip_runtime.h>
#include <hip/hip_bf16.h>

#define B_ 4096
#define D_ 512
#define G_ 128
#define N_ (G_ * G_)
#define LDA_ (D_ + 4)   // padded LDS row stride (floats) -> conflict-free across 16 lanes
#define KC_ 64          // K-chunk staged per wave per step
#define LDW_ (KC_ + 4)  // padded W-chunk row stride (floats), 68 mod 64 = 4 -> conflict-free
#define THR_ 0.95f
#define AS3 __attribute__((address_space(3)))

typedef __attribute__((ext_vector_type(2))) float v2f;
typedef __attribute__((ext_vector_type(8))) float v8f;

// ---------------------------------------------------------------------------
// Kernel 0: schedule scalars (epoch / max_epochs live in device memory)
// ---------------------------------------------------------------------------
__global__ void som_sched_kernel(const int* __restrict__ epoch,
                                 const int* __restrict__ max_epochs,
                                 float* __restrict__ wsf) {
    float me = (float)max_epochs[0];
    float p = (me > 0.0f) ? ((me - (float)epoch[0]) / me) : 0.0f;
    p = fminf(1.0f, fmaxf(0.0f, p));
    float p4 = p * p * p * p;
    int ctx = (int)(p4 * 2.0f);              // CTX_MIN + int(p^4*(CTX_MAX-CTX_MIN))
    float lr = 0.05f + p * (0.2f - 0.05f);   // LR_MIN + p*(LR_MAX-LR_MIN)
    wsf[0] = 0.0f;        // running sum of min_dists
    wsf[1] = lr;
    wsf[2] = (float)ctx;
}

// ---------------------------------------------------------------------------
// Kernel 1: zero numerator accumulator (lives in d_out som region) + denom
// ---------------------------------------------------------------------------
__global__ void som_zero_kernel(float* __restrict__ acc, float* __restrict__ denom) {
    size_t i = (size_t)blockIdx.x * blockDim.x + threadIdx.x;
    if (i < (size_t)N_ * D_) acc[i] = 0.0f;
    if (i < (size_t)N_)      denom[i] = 0.0f;
}

// ---------------------------------------------------------------------------
// Kernel 2: per-cell squared norms of SOM vectors
// ---------------------------------------------------------------------------
__global__ void som_wnorm_kernel(const float* __restrict__ som, float* __restrict__ wnorm) {
    int n = blockIdx.x * blockDim.x + threadIdx.x;
    if (n >= N_) return;
    const float4* p = (const float4*)(som + (size_t)n * D_);
    float s = 0.0f;
    #pragma unroll 4
    for (int k = 0; k < D_ / 4; ++k) {
        float4 v = p[k];
        s += v.x * v.x + v.y * v.y + v.z * v.z + v.w * v.w;
    }
    wnorm[n] = s;
}

// ---------------------------------------------------------------------------
// Kernel 3: fused distance GEMM (WMMA f32 16x16x4) + argmin + class-argmin
//   grid = B/16 blocks, 256 threads (8 waves). Block owns 16 rows of A in LDS;
//   each wave sweeps 128 of the 1024 16-column W tiles, staging W K-chunks
//   into its own double-buffered LDS slab via global_load_async_to_lds_b128,
//   reading them back with ds_load through explicit addrspace(3) pointers.
// ---------------------------------------------------------------------------
__global__ __launch_bounds__(256) void som_bmu_kernel(
    const float* __restrict__ act, const int* __restrict__ labels,
    const float* __restrict__ som, const int* __restrict__ cell_labels,
    const float* __restrict__ wnorm, float* __restrict__ ws_sum,
    int* __restrict__ bmu_out, int* __restrict__ cbmu_out) {

    __shared__ float Alds[16 * LDA_];
    __shared__ float xnorm[16];
    __shared__ int   lab[16];
    __shared__ float red_s[8][16];
    __shared__ int   red_i[8][16];
    __shared__ float redc_s[8][16];
    __shared__ int   redc_i[8][16];
    extern __shared__ float dynlds[];        // W chunk buffers: [8 waves][2][16*LDW_]

    const int tid  = threadIdx.x;
    const int wave = tid >> 5;
    const int lane = tid & 31;
    const int row0 = blockIdx.x * 16;

    // Cooperative A-tile load: 16 rows x 512 floats (row stride padded to 516)
    for (int i = tid; i < 16 * (D_ / 4); i += 256) {
        int r  = i >> 7;          // /128 float4s per row
        int c4 = i & 127;
        ((float4*)(Alds + r * LDA_))[c4] =
            ((const float4*)(act + (size_t)(row0 + r) * D_))[c4];
    }
    if (tid < 16) lab[tid] = labels[row0 + tid];
    __syncthreads();
    if (tid < 16) {
        float s = 0.0f;
        for (int k = 0; k < D_; ++k) { float v = Alds[tid * LDA_ + k]; s += v * v; }
        xnorm[tid] = s;
    }
    __syncthreads();

    const int nlane = lane & 15;   // M index for A frag / N index for B frag
    const int half  = lane >> 4;   // lane group -> K offset
    const int koff  = half * 2;

    int mylab[8];
    #pragma unroll
    for (int r = 0; r < 8; ++r) mylab[r] = lab[r + 8 * half];

    float best[8], cbest[8];
    int   bidx[8], cbidx[8];
    #pragma unroll
    for (int r = 0; r < 8; ++r) {
        best[r] = 3.0e38f;  bidx[r]  = 0x7fffffff;
        cbest[r] = 3.0e38f; cbidx[r] = 0x7fffffff;
    }

    // per-wave double buffer as 32-bit LDS byte offsets (generic->LDS is a
    // low-32-bit truncation; keeps buffer select in 32-bit, enables ds_load)
    const unsigned wbase0 = (unsigned)(size_t)dynlds + (unsigned)(wave * 2 * 16 * LDW_ * 4);
    const unsigned wbase1 = wbase0 + (unsigned)(16 * LDW_ * 4);

    // per-lane fixed mapping for async chunk loads:
    //   instruction j covers rows 2j,(2j+1); lane -> row = 2j + (lane>>4),
    //   cols = (lane&15)*4 .. +3  (16B per lane, two 256B coalesced segments)
    const int arow_off = (lane >> 4);
    const int acol     = (lane & 15) * 4;
    const unsigned lds_lane_off = (unsigned)((arow_off * LDW_ + acol) * 4);

    const float* arow = Alds + nlane * LDA_ + koff;
    const unsigned brow_off = (unsigned)((nlane * LDW_ + koff) * 4);

    for (int t = wave; t < N_ / 16; t += 8) {          // uniform trip count
        const int n0 = t * 16;
        const int n  = n0 + nlane;

        // issue chunk 0 into buffer 0
        {
            const float* g = som + (size_t)(n0 + arow_off) * D_ + acol;
            unsigned l = wbase0 + lds_lane_off;
            #pragma unroll
            for (int j = 0; j < 8; ++j) {
                asm volatile("global_load_async_to_lds_b128 %0, %1, off"
                             :: "v"(l), "v"(g) : "memory");
                g += 2 * D_;
                l += 2 * LDW_ * 4;
            }
        }

        v8f c = {};
        #pragma unroll 1
        for (int kc = 0; kc < D_; kc += KC_) {
            const int cur = (kc >> 6) & 1;
            if (kc + KC_ < D_) {
                // prefetch next chunk into the other buffer
                const float* g = som + (size_t)(n0 + arow_off) * D_ + (kc + KC_) + acol;
                unsigned l = (cur ? wbase0 : wbase1) + lds_lane_off;
                #pragma unroll
                for (int j = 0; j < 8; ++j) {
                    asm volatile("global_load_async_to_lds_b128 %0, %1, off"
                                 :: "v"(l), "v"(g) : "memory");
                    g += 2 * D_;
                    l += 2 * LDW_ * 4;
                }
                asm volatile("s_wait_asynccnt 8" ::: "memory");
            } else {
                asm volatile("s_wait_asynccnt 0" ::: "memory");
            }
            const AS3 float* brow =
                (const AS3 float*)(size_t)((cur ? wbase1 : wbase0) + brow_off);
            #pragma unroll
            for (int kk = 0; kk < KC_; kk += 4) {
                v2f a = *(const v2f*)(arow + kc + kk);      // A: M=nlane, K=koff..+1
                v2f b = *(const AS3 v2f*)(brow + kk);       // B: N=nlane, K=koff..+1 (ds_load)
                c = __builtin_amdgcn_wmma_f32_16x16x4_f32(
                        false, a, false, b, (short)0, c, false, false);
            }
        }

        float wn = wnorm[n];
        int   cl = cell_labels[n];
        #pragma unroll
        for (int r = 0; r < 8; ++r) {                  // c[r]: M = r + 8*half, N = nlane
            float s = fmaf(-2.0f, c[r], wn);
            bool bt = s < best[r];                     // branchless: cndmask, no exec
            bidx[r] = bt ? n : bidx[r];
            best[r] = bt ? s : best[r];
            bool ct = (cl == mylab[r]) & (s < cbest[r]);
            cbidx[r] = ct ? n : cbidx[r];
            cbest[r] = ct ? s : cbest[r];
        }
    }

    // Reduce across the 16 lanes of each half (same rows, different columns)
    #pragma unroll
    for (int r = 0; r < 8; ++r) {
        float v = best[r];  int ix = bidx[r];
        float cv = cbest[r]; int cix = cbidx[r];
        #pragma unroll
        for (int off = 1; off < 16; off <<= 1) {
            float ov  = __shfl_xor(v, off, 32);
            int   oi  = __shfl_xor(ix, off, 32);
            bool  tk  = (ov < v) || (ov == v && oi < ix);
            v  = tk ? ov : v;   ix  = tk ? oi : ix;
            float ocv = __shfl_xor(cv, off, 32);
            int   oci = __shfl_xor(cix, off, 32);
            bool  tkc = (ocv < cv) || (ocv == cv && oci < cix);
            cv = tkc ? ocv : cv; cix = tkc ? oci : cix;
        }
        if (nlane == 0) {
            int m = r + 8 * half;
            red_s[wave][m]  = v;  red_i[wave][m]  = ix;
            redc_s[wave][m] = cv; redc_i[wave][m] = cix;
        }
    }
    __syncthreads();

    // Reduce across the 8 waves; one thread per row finalizes
    if (tid < 16) {
        float v = red_s[0][tid];  int ix = red_i[0][tid];
        float cv = redc_s[0][tid]; int cix = redc_i[0][tid];
        #pragma unroll
        for (int w = 1; w < 8; ++w) {
            float ov = red_s[w][tid];  int oi = red_i[w][tid];
            bool  tk = (ov < v) || (ov == v && oi < ix);
            v  = tk ? ov : v;   ix  = tk ? oi : ix;
            float ocv = redc_s[w][tid]; int oci = redc_i[w][tid];
            bool  tkc = (ocv < cv) || (ocv == cv && oci < cix);
            cv = tkc ? ocv : cv; cix = tkc ? oci : cix;
        }
        float md = xnorm[tid] + v;          // dist = ||x||^2 + (||w||^2 - 2 x.w)
        if (md < 0.0f) md = 0.0f;           // clip
        atomicAdd(ws_sum, md);
        bmu_out[row0 + tid]  = ix;
        cbmu_out[row0 + tid] = (cix == 0x7fffffff) ? 0 : cix;
    }
}

// ---------------------------------------------------------------------------
// Kernel 4: som_errors = 0.01 * rel * (w[class_bmu] - x) * (rel >= THR)
// ---------------------------------------------------------------------------
__global__ __launch_bounds__(256) void som_errors_kernel(
    const float* __restrict__ act, const float* __restrict__ som,
    const float* __restrict__ rel_in, const int* __restrict__ cbmu,
    float* __restrict__ err) {
    int b  = blockIdx.x;
    int cb = cbmu[b];
    float rel   = rel_in[cb] * 0.01f;                   // /100
    float scale = (rel >= THR_) ? 0.01f * rel : 0.0f;
    const float* w = som + (size_t)cb * D_;
    const float* a = act + (size_t)b * D_;
    float* o       = err + (size_t)b * D_;
    for (int d = threadIdx.x; d < D_; d += 256)
        o[d] = scale * (w[d] - a[d]);
}

// ---------------------------------------------------------------------------
// Kernel 5: sparse neighborhood scatter (powers has <= (2ctx+1)^2 nnz per row)
//   cell index n = i*G + j with i ~ bmu%G (y), j ~ bmu/G (x)  [reference quirk]
// ---------------------------------------------------------------------------
__global__ __launch_bounds__(256) void som_scatter_kernel(
    const float* __restrict__ act, const int* __restrict__ bmu,
    const float* __restrict__ wsf, float* __restrict__ acc,
    float* __restrict__ denom) {
    const int b   = blockIdx.x;
    const int tid = threadIdx.x;
    const int best = bmu[b];
    const int bx = best / G_;      // compared against axis-1 (j)
    const int by = best % G_;      // compared against axis-0 (i)
    const float lr = wsf[1];
    const int  ctx = (int)wsf[2];
    const float* src = act + (size_t)b * D_;
    for (int di = -ctx; di <= ctx; ++di) {
        int i = by + di; if (i < 0 || i >= G_) continue;
        for (int dj = -ctx; dj <= ctx; ++dj) {
            int j = bx + dj; if (j < 0 || j >= G_) continue;
            int cheb = max(abs(di), abs(dj));
            float p = lr / (float)(1 << cheb);
            int cell = i * G_ + j;
            float* dst = acc + (size_t)cell * D_;
            for (int d = tid; d < D_; d += 256) atomicAdd(dst + d, p * src[d]);
            if (tid == 0) atomicAdd(denom + cell, p);
        }
    }
}

// ---------------------------------------------------------------------------
// Kernel 6: new_som = gate ? som + numerator - som*denom : som   (in place:
//   out_som currently holds the accumulated numerator)
// ---------------------------------------------------------------------------
__global__ void som_finalize_kernel(const float* __restrict__ som,
                                    const float* __restrict__ denom,
                                    const float* __restrict__ wsf,
                                    float* __restrict__ out_som) {
    size_t i = (size_t)blockIdx.x * blockDim.x + threadIdx.x;
    if (i >= (size_t)N_ * D_) return;
    bool gate = (wsf[0] * (1.0f / (float)B_)) > 1e-4f;
    float sv = som[i];
    float nu = out_som[i];
    float dn = denom[i / D_];
    out_som[i] = gate ? (sv + nu - sv * dn) : sv;
}

// ---------------------------------------------------------------------------
extern "C" void kernel_launch(void* const* d_in, const int* in_sizes, int n_in,
                              void* d_out, int out_size, void* d_ws, size_t ws_size,
                              hipStream_t stream) {
    const float* act         = (const float*)d_in[0];
    const int*   labels      = (const int*)d_in[1];
    const float* som         = (const float*)d_in[2];
    const int*   cell_labels = (const int*)d_in[3];
    const float* rel         = (const float*)d_in[4];
    const int*   epoch       = (const int*)d_in[5];
    const int*   max_epochs  = (const int*)d_in[6];

    float* out     = (float*)d_out;
    float* err     = out;                        // (B, D) som_errors
    float* out_som = out + (size_t)B_ * D_;      // (G, G, D) new_som

    float* wsf   = (float*)d_ws;                 // [0]=sum, [1]=lr, [2]=ctx
    float* wnorm = wsf + 16;                     // N_ floats
    float* denom = wnorm + N_;                   // N_ floats
    int*   bmu   = (int*)(denom + N_);           // B_ ints
    int*   cbmu  = bmu + B_;                     // B_ ints

    som_sched_kernel<<<1, 1, 0, stream>>>(epoch, max_epochs, wsf);

    {   size_t tot = (size_t)N_ * D_;
        int blocks = (int)((tot + 255) / 256);
        som_zero_kernel<<<blocks, 256, 0, stream>>>(out_som, denom);
    }

    som_wnorm_kernel<<<N_ / 256, 256, 0, stream>>>(som, wnorm);

    // dynamic LDS: 8 waves x 2 buffers x 16 rows x LDW_ floats
    size_t dyn_bytes = (size_t)8 * 2 * 16 * LDW_ * sizeof(float);
    som_bmu_kernel<<<B_ / 16, 256, dyn_bytes, stream>>>(act, labels, som, cell_labels,
                                                        wnorm, wsf, bmu, cbmu);

    som_errors_kernel<<<B_, 256, 0, stream>>>(act, som, rel, cbmu, err);

    som_scatter_kernel<<<B_, 256, 0, stream>>>(act, bmu, wsf, out_som, denom);

    {   size_t tot = (size_t)N_ * D_;
        int blocks = (int)((tot + 255) / 256);
        som_finalize_kernel<<<blocks, 256, 0, stream>>>(som, denom, wsf, out_som);
    }
}